// Prototype_Balanced_Contrastive_Loss_old_33414845562972
// MI455X (gfx1250) — compile-verified
//
#include <hip/hip_runtime.h>
#include <hip/hip_bf16.h>
#include <stdint.h>

// ---------------- problem constants (match reference setup_inputs) ----------
#define B_    8
#define C_    256
#define H_    128
#define W_    128
#define HW_   (H_ * W_)
#define H0_   512
#define W0_   512
#define NC_   21      // num_class + 1
#define NOC_  15      // num_old_class
#define NOCH_ 16      // outputs_old channels (num_old_class + 1)
#define THRESH_ 0.5f
#define INV_TAU_ 10.0f

// Gram matrix geometry: rows 0..167 = Sb (class*8+batch), 168..188 = P, 189..191 = 0
#define MROWS_ 192
#define MTILES_ 12            // 192/16
#define LDS_PITCH_ 257        // 257 % 64 == 1 -> conflict-free column gathers

// ---------------- workspace layout (bytes) ----------------------------------
// [S(21*8*256 f32) | T(21*8*256 f32) | cnt_s(168 u32) | cnt_t(168 u32)]  <- zeroed
// [MN(192*256 f32) | TN(168*256 f32) | G(192*192 f32) | pl(u8 B*HW) | pl_old(u8 B*HW)]
#define OFF_S      0
#define OFF_T      (OFF_S + NC_*B_*C_*4)          // 172032
#define OFF_CNTS   (OFF_T + NC_*B_*C_*4)          // 344064
#define OFF_CNTT   (OFF_CNTS + NC_*B_*4)          // 344736
#define ZERO_DW    ((OFF_CNTT + NC_*B_*4) / 4)    // 86352 dwords zeroed per call
#define OFF_MN     (OFF_CNTT + NC_*B_*4)          // 345408
#define OFF_TN     (OFF_MN + MROWS_*C_*4)         // 542016
#define OFF_G      (OFF_TN + NC_*B_*C_*4)         // 714048
#define OFF_PL     (OFF_G + MROWS_*MROWS_*4)      // 861504
#define OFF_PLO    (OFF_PL + B_*HW_)              // 992576
// total ws use: 1123648 bytes

typedef float v2f __attribute__((ext_vector_type(2)));
typedef float v8f __attribute__((ext_vector_type(8)));

// ---------------- kernel 0: zero accumulators --------------------------------
__global__ void k_init(uint32_t* __restrict__ p, int n) {
  int i = blockIdx.x * 256 + threadIdx.x;
  if (i < n) p[i] = 0u;
}

// ---------------- kernel 1: pseudo-labels + per-(class,batch) counts ---------
__global__ __launch_bounds__(256) void k_labels(
    const int* __restrict__ labels, const float* __restrict__ outputs_old,
    uint8_t* __restrict__ pl, uint8_t* __restrict__ plo,
    uint32_t* __restrict__ cnt_s, uint32_t* __restrict__ cnt_t) {
  __shared__ uint32_t hs[NC_], ht[NC_];
  const int t = threadIdx.x;
  if (t < NC_) { hs[t] = 0u; ht[t] = 0u; }
  __syncthreads();
  const int gid = blockIdx.x * 256 + t;       // 256 | HW -> whole block same b
  const int b = gid / HW_;
  const int p = gid % HW_;
  const int h = p / W_, w = p % W_;
  const size_t pix = (size_t)(4 * h) * W0_ + (size_t)(4 * w);
  const int lab = labels[(size_t)b * H0_ * W0_ + pix];
  int po = 0;
  if (lab == 0) {  // argmax (first max) over thresholded old logits
    const float* op = outputs_old + (size_t)b * NOCH_ * H0_ * W0_ + pix;
    float v = op[0]; float best = (v < THRESH_) ? 0.0f : v;
    #pragma unroll
    for (int ch = 1; ch < NOCH_; ++ch) {
      v = op[(size_t)ch * H0_ * W0_];
      v = (v < THRESH_) ? 0.0f : v;
      if (v > best) { best = v; po = ch; }
    }
  }
  const int plv = lab + po;                   // po==0 when lab!=0
  pl[gid]  = (uint8_t)plv;
  plo[gid] = (uint8_t)po;
  atomicAdd(&hs[plv], 1u);
  atomicAdd(&ht[po], 1u);
  __syncthreads();
  if (t < NC_) {
    if (hs[t]) atomicAdd(&cnt_s[t * B_ + b], hs[t]);
    if (ht[t]) atomicAdd(&cnt_t[t * B_ + b], ht[t]);
  }
}

// ---------------- kernel 2: HBM-bound segmented sum  S[k][b][c] += feat ------
// grid = (HW/4096, C/64, B), block = 256 (8 waves); float4 stream + ds_add_f32
__global__ __launch_bounds__(256) void k_segsum(
    const float* __restrict__ feat, const uint8_t* __restrict__ cls,
    float* __restrict__ out /* [NC][B][C] */) {
  const int CCH = 64, HWCH = 4096;
  __shared__ float acc[NC_ * 64];
  __shared__ uint8_t kls[4096];
  const int t = threadIdx.x;
  const int b = blockIdx.z, cbase = blockIdx.y * CCH, hwb = blockIdx.x * HWCH;
  for (int i = t; i < NC_ * CCH; i += 256) acc[i] = 0.0f;
  const uint32_t* kp = (const uint32_t*)(cls + (size_t)b * HW_ + hwb);
  uint32_t* ks = (uint32_t*)kls;
  for (int i = t; i < HWCH / 4; i += 256) ks[i] = kp[i];
  __syncthreads();
  const int wave = t >> 5, lane = t & 31;
  for (int ci = wave; ci < CCH; ci += 8) {
    const float4* fp =
        (const float4*)(feat + ((size_t)(b * C_ + cbase + ci)) * HW_ + hwb);
    for (int j = lane; j < HWCH / 4; j += 32) {
      __builtin_prefetch((const void*)(fp + j + 64), 0, 0);  // global_prefetch_b8
      const float4 v = fp[j];
      const uint32_t kk = ks[j];                // 4 packed class ids
      atomicAdd(&acc[(kk & 255u) * CCH + ci], v.x);
      atomicAdd(&acc[((kk >> 8) & 255u) * CCH + ci], v.y);
      atomicAdd(&acc[((kk >> 16) & 255u) * CCH + ci], v.z);
      atomicAdd(&acc[(kk >> 24) * CCH + ci], v.w);
    }
  }
  __syncthreads();
  for (int i = t; i < NC_ * CCH; i += 256) {
    const int k = i / CCH, ci = i % CCH;
    const float v = acc[i];
    if (v != 0.0f)
      atomicAdd(&out[((size_t)k * B_ + b) * C_ + cbase + ci], v);
  }
}

// ---------------- kernel 3a: means + L2 normalize -> MN (Sb;P;0), TN ---------
// 360 waves: rows 0..167 Sb, 168..188 P, 189..191 zero, 192..359 Tb
__global__ __launch_bounds__(256) void k_norm(
    const float* __restrict__ S, const float* __restrict__ T,
    const uint32_t* __restrict__ cnt_s, const uint32_t* __restrict__ cnt_t,
    const float* __restrict__ protos, float* __restrict__ MN,
    float* __restrict__ TN) {
  const int wave = threadIdx.x >> 5, lane = threadIdx.x & 31;
  const int row = blockIdx.x * 8 + wave;
  if (row >= 360) return;
  float v[8];
  float* dst;
  if (row < 168) {
    const float inv = 1.0f / fmaxf((float)cnt_s[row], 1.0f);
    #pragma unroll
    for (int i = 0; i < 8; ++i) v[i] = S[(size_t)row * C_ + lane + i * 32] * inv;
    dst = MN + (size_t)row * C_;
  } else if (row < 189) {
    const int pr = row - 168;
    #pragma unroll
    for (int i = 0; i < 8; ++i) v[i] = protos[(size_t)pr * C_ + lane + i * 32];
    dst = MN + (size_t)row * C_;
  } else if (row < MROWS_) {
    #pragma unroll
    for (int i = 0; i < 8; ++i) v[i] = 0.0f;
    dst = MN + (size_t)row * C_;
  } else {
    const int r = row - MROWS_;
    const float inv = 1.0f / fmaxf((float)cnt_t[r], 1.0f);
    #pragma unroll
    for (int i = 0; i < 8; ++i) v[i] = T[(size_t)r * C_ + lane + i * 32] * inv;
    dst = TN + (size_t)r * C_;
  }
  float ss = 0.0f;
  #pragma unroll
  for (int i = 0; i < 8; ++i) ss += v[i] * v[i];
  #pragma unroll
  for (int o = 16; o >= 1; o >>= 1) ss += __shfl_xor(ss, o, 32);
  const float inv = 1.0f / fmaxf(sqrtf(ss), 1e-12f);
  #pragma unroll
  for (int i = 0; i < 8; ++i) dst[lane + i * 32] = v[i] * inv;
}

// ---------------- kernel 3b: G = MN * MN^T via V_WMMA_F32_16X16X4_F32 --------
// One workgroup, 8 waves. MN (192x256) staged in dynamic LDS (192x257 f32,
// ~193 KB < 320 KB WGP LDS). 144 output tiles / 8 waves = 18 tiles/wave,
// each tile = 64 chained wmma ops over K=256.
__global__ __launch_bounds__(256) void k_gram(const float* __restrict__ MN,
                                              float* __restrict__ G) {
  extern __shared__ float lds[];  // [MROWS_][LDS_PITCH_]
  const int t = threadIdx.x;
  for (int i = t; i < MROWS_ * C_; i += 256) {
    const int r = i >> 8, c = i & 255;
    lds[r * LDS_PITCH_ + c] = MN[i];
  }
  __syncthreads();
  const int wave = t >> 5, lane = t & 31;
  const int rloc = lane & 15;          // row within 16-row tile
  const int ksel = (lane >> 4) * 2;    // lanes 0-15: K+0/K+1, lanes 16-31: K+2/K+3
  for (int tile = wave; tile < MTILES_ * MTILES_; tile += 8) {
    const int ti = tile / MTILES_, tj = tile % MTILES_;
    const float* arow = lds + (ti * 16 + rloc) * LDS_PITCH_ + ksel;
    const float* brow = lds + (tj * 16 + rloc) * LDS_PITCH_ + ksel;  // B = MN^T slice
    v8f acc = {0.f, 0.f, 0.f, 0.f, 0.f, 0.f, 0.f, 0.f};
    for (int k = 0; k < C_; k += 4) {
      v2f a = {arow[k], arow[k + 1]};
      v2f b = {brow[k], brow[k + 1]};
      acc = __builtin_amdgcn_wmma_f32_16x16x4_f32(
          /*neg_a=*/false, a, /*neg_b=*/false, b,
          /*c_mod=*/(short)0, acc, /*reuse_a=*/false, /*reuse_b=*/false);
    }
    // C/D layout: VGPR r -> (M = r + 8*(lane>=16), N = lane&15)
    const int m0 = ti * 16 + ((lane >> 4) << 3);
    const int n = tj * 16 + (lane & 15);
    #pragma unroll
    for (int r = 0; r < 8; ++r)
      G[(size_t)(m0 + r) * MROWS_ + n] = acc[r];
  }
}

// ---------------- kernel 3c: contrastive terms -> scalar loss ----------------
__global__ __launch_bounds__(256) void k_final(
    const float* __restrict__ G, const float* __restrict__ MN,
    const float* __restrict__ TN, const uint32_t* __restrict__ cnt_s,
    const uint32_t* __restrict__ cnt_t, float* __restrict__ out) {
  __shared__ float red[256];
  __shared__ float csum[NC_], tsum[NC_], lossS;
  __shared__ uint8_t presS[NC_ * B_], presT[NC_ * B_];
  const int t = threadIdx.x;
  if (t < NC_ * B_) {
    const int k = t / B_;
    presS[t] = (uint8_t)((cnt_s[t] > 0u) && (k != 0));
    presT[t] = (uint8_t)((cnt_t[t] > 0u) && (k != 0));
  }
  __syncthreads();
  if (t < NC_) {
    float s = 1.0f, s2 = 1.0f;
    for (int b = 0; b < B_; ++b) { s += presS[t * B_ + b]; s2 += presT[t * B_ + b]; }
    csum[t] = s;   // cnt_s (pres sum + 1)
    tsum[t] = s2;  // cnt_t
  }
  __syncthreads();
  float contrib = 0.0f, nexist = 0.0f;
  if (t < NC_ * B_) {
    const int i = t / B_, b = t % B_;
    const int row = i * 8 + b;
    if (presS[t] && i >= 1 && i <= NOC_) {
      float neg = 0.0f;
      for (int m = 1; m < NC_; ++m) {
        if (m == i) continue;
        float s = 0.0f;
        for (int n = 0; n < B_; ++n)
          if (presS[m * B_ + n]) s += expf(G[(size_t)row * MROWS_ + m * 8 + n] * INV_TAU_);
        s += expf(G[(size_t)row * MROWS_ + 168 + m] * INV_TAU_);  // vs prototype m
        neg += s / csum[m];
      }
      float sum_k = 0.0f;
      const float* sb = MN + (size_t)row * C_;
      for (int k = 0; k < B_; ++k) {
        if (!presT[i * B_ + k]) continue;
        const float* tb = TN + (size_t)(i * 8 + k) * C_;
        float d = 0.0f;
        for (int c = 0; c < C_; ++c) d += sb[c] * tb[c];
        d *= INV_TAU_;
        sum_k += logf(neg + expf(d)) - d;
      }
      const float dP = G[(size_t)row * MROWS_ + 168 + i] * INV_TAU_;
      sum_k += logf(neg + expf(dP)) - dP;
      contrib = sum_k / (tsum[i] * fmaxf(csum[i] - 1.0f, 1.0f));
    }
    if (b == 0 && i >= 1 && i <= NOC_ && csum[i] > 1.0f) nexist = 1.0f;
  }
  red[t] = contrib;
  __syncthreads();
  for (int s = 128; s > 0; s >>= 1) {
    if (t < s) red[t] += red[t + s];
    __syncthreads();
  }
  if (t == 0) lossS = red[0];
  __syncthreads();
  red[t] = nexist;
  __syncthreads();
  for (int s = 128; s > 0; s >>= 1) {
    if (t < s) red[t] += red[t + s];
    __syncthreads();
  }
  if (t == 0) {
    const float n = red[0];
    out[0] = (n > 0.0f) ? lossS / n : lossS;
  }
}

// ---------------- launch ------------------------------------------------------
extern "C" void kernel_launch(void* const* d_in, const int* in_sizes, int n_in,
                              void* d_out, int out_size, void* d_ws,
                              size_t ws_size, hipStream_t stream) {
  const int* labels = (const int*)d_in[0];
  const float* features_old = (const float*)d_in[1];
  const float* features = (const float*)d_in[2];
  const float* outputs_old = (const float*)d_in[3];
  const float* prototypes = (const float*)d_in[4];

  char* ws = (char*)d_ws;
  float* S = (float*)(ws + OFF_S);
  float* T = (float*)(ws + OFF_T);
  uint32_t* cnt_s = (uint32_t*)(ws + OFF_CNTS);
  uint32_t* cnt_t = (uint32_t*)(ws + OFF_CNTT);
  float* MN = (float*)(ws + OFF_MN);
  float* TN = (float*)(ws + OFF_TN);
  float* G = (float*)(ws + OFF_G);
  uint8_t* pl = (uint8_t*)(ws + OFF_PL);
  uint8_t* plo = (uint8_t*)(ws + OFF_PLO);

  k_init<<<(ZERO_DW + 255) / 256, 256, 0, stream>>>((uint32_t*)ws, ZERO_DW);
  k_labels<<<(B_ * HW_) / 256, 256, 0, stream>>>(labels, outputs_old, pl, plo,
                                                 cnt_s, cnt_t);
  dim3 g2(HW_ / 4096, C_ / 64, B_);  // (4,4,8)
  k_segsum<<<g2, 256, 0, stream>>>(features, pl, S);
  k_segsum<<<g2, 256, 0, stream>>>(features_old, plo, T);
  k_norm<<<45, 256, 0, stream>>>(S, T, cnt_s, cnt_t, prototypes, MN, TN);
  k_gram<<<1, 256, MROWS_ * LDS_PITCH_ * sizeof(float), stream>>>(MN, G);
  k_final<<<1, 256, 0, stream>>>(G, MN, TN, cnt_s, cnt_t, (float*)d_out);
}